// BDH_Reasoner_19198503813240
// MI455X (gfx1250) — compile-verified
//
#include <hip/hip_runtime.h>
#include <math.h>

// ---------------------------------------------------------------------------
// BDH reasoner step for MI455X (gfx1250, wave32).
// Bandwidth-bound: ~900MB of HBM traffic @ 23.3 TB/s => ~38us floor.
//   K1 enc:     h = relu(W_enc @ x + b)            (128MB stream, VALU fp32)
//   K2 topk:    exact radix-select kth of h, build h_sparse, ||h||^2
//   K3 syn:     y = h_sparse @ S  AND  ||S||^2 fused in ONE 256MB stream,
//               using V_WMMA_F32_16X16X4_F32 (f32-exact) + NT loads
//   K4 finish:  new_state=tanh(y+state), logits, norm^2 via
//               ||S+c*hh'||^2 = ||S||^2 + 2c*(h'Sh) + c^2*||h||^4  (no extra pass)
//   K5 update:  new_syn = (S + c*h[r]*h[c]) * inv_norm  (256MB rd + 256MB wr, NT)
// Deterministic: integer LDS atomics only; all float reductions fixed-order.
// ---------------------------------------------------------------------------

#define INPUT_DIM 4096
#define HIDDEN    8192
#define TOPK      409
#define HEBB_LR   0.001f

typedef __attribute__((ext_vector_type(2))) float v2f;
typedef __attribute__((ext_vector_type(4))) float v4f;
typedef __attribute__((ext_vector_type(8))) float v8f;

// workspace layout (float offsets)
#define WS_H      0                         // h[8192]
#define WS_HS     8192                      // h_sparse[8192]
#define WS_YPART  16384                     // y partials [8 chunks][8192]
#define WS_SSQ    (16384 + 8*8192)          // per-block ||S||^2 partials [512]
#define WS_SCAL   (WS_SSQ + 512)            // [0]=kth [1]=||h||^2 [2]=inv_norm

// ---------------- K1: encoder matvec, one wave per output row --------------
__global__ __launch_bounds__(256) void enc_kernel(const float* __restrict__ x,
                                                  const float* __restrict__ W,
                                                  const float* __restrict__ b,
                                                  float* __restrict__ ws) {
  const int wave = threadIdx.x >> 5, lane = threadIdx.x & 31;
  const int r = blockIdx.x * 8 + wave;
  const v4f* __restrict__ W4 = (const v4f*)(W + (size_t)r * INPUT_DIM);
  const v4f* __restrict__ x4 = (const v4f*)x;
  float s = 0.f;
#pragma unroll 4
  for (int i = 0; i < INPUT_DIM / 4 / 32; ++i) {
    const int idx = i * 32 + lane;
    v4f wv = __builtin_nontemporal_load(W4 + idx);   // W streamed once, NT
    v4f xv = x4[idx];
    s += wv.x * xv.x + wv.y * xv.y + wv.z * xv.z + wv.w * xv.w;
  }
  for (int m = 16; m; m >>= 1) s += __shfl_xor(s, m, 32);
  if (lane == 0) ws[WS_H + r] = fmaxf(s + b[r], 0.f);
}

// ---------------- K2: exact kth-largest via bitwise radix select -----------
__global__ __launch_bounds__(1024) void topk_kernel(float* __restrict__ ws) {
  __shared__ unsigned sh[HIDDEN];
  __shared__ int cnt;
  __shared__ float red[1024];
  const int tid = threadIdx.x;
  for (int i = tid; i < HIDDEN; i += 1024)
    sh[i] = __float_as_uint(ws[WS_H + i]);           // h >= 0: bits monotone
  __syncthreads();
  unsigned prefix = 0u;
  for (int bit = 31; bit >= 0; --bit) {
    const unsigned cand = prefix | (1u << bit);
    if (tid == 0) cnt = 0;
    __syncthreads();
    int local = 0;
    for (int i = tid; i < HIDDEN; i += 1024) local += (sh[i] >= cand) ? 1 : 0;
    atomicAdd(&cnt, local);                          // integer: deterministic
    __syncthreads();
    if (cnt >= TOPK) prefix = cand;                  // max t with count(>=t)>=K
    __syncthreads();
  }
  const float kth = __uint_as_float(prefix);
  float ssq = 0.f;
  for (int i = tid; i < HIDDEN; i += 1024) {
    const float v = __uint_as_float(sh[i]);
    const float hs = (v >= kth) ? v : 0.f;           // keep ties, like reference
    ws[WS_HS + i] = hs;
    ssq += hs * hs;
  }
  red[tid] = ssq;
  __syncthreads();
  for (int s = 512; s; s >>= 1) {
    if (tid < s) red[tid] += red[tid + s];
    __syncthreads();
  }
  if (tid == 0) { ws[WS_SCAL + 0] = kth; ws[WS_SCAL + 1] = red[0]; }
}

// ---------------- K3: fused  y = h@S  +  ||S||^2  (WMMA f32 16x16x4) -------
// wave gw: column tile c0 = (gw>>3)*16, row chunk = gw&7 (1024 rows each).
// A[m,k] = h[k0+k] broadcast over m  => D[m,n] = y_partial[c0+n] for all m.
__global__ __launch_bounds__(256) void syn_pass_kernel(const float* __restrict__ S,
                                                       float* __restrict__ ws) {
  const int wave = threadIdx.x >> 5, lane = threadIdx.x & 31;
  const int gw = blockIdx.x * 8 + wave;        // 0..4095
  const int c0 = (gw >> 3) * 16;               // column tile base
  const int chunk = gw & 7;                    // row chunk
  const int n = lane & 15;
  const int khalf = lane >> 4;                 // lanes 0-15: K{0,1}; 16-31: K{2,3}
  const int kbase = chunk * 1024;
  const float* __restrict__ hs = ws + WS_HS;
  v8f acc = {};
  float ssq = 0.f;
  for (int k0 = kbase; k0 < kbase + 1024; k0 += 4) {
    const int ka = k0 + 2 * khalf;
    v2f a = *(const v2f*)(hs + ka);            // A: h[ka], h[ka+1]
    const float b0 = __builtin_nontemporal_load(S + (size_t)ka * HIDDEN + c0 + n);
    const float b1 = __builtin_nontemporal_load(S + (size_t)(ka + 1) * HIDDEN + c0 + n);
    v2f bm = {b0, b1};                         // B: S[ka, c0+n], S[ka+1, c0+n]
    ssq += b0 * b0 + b1 * b1;                  // fused Frobenius accumulation
    acc = __builtin_amdgcn_wmma_f32_16x16x4_f32(false, a, false, bm,
                                                (short)0, acc, false, false);
  }
  // D VGPR0 lanes 0-15 = D[M=0, N=lane] = y_partial[c0+lane]
  if (lane < 16) ws[WS_YPART + chunk * HIDDEN + c0 + lane] = acc[0];
  // deterministic block reduce of ssq
  for (int m = 16; m; m >>= 1) ssq += __shfl_xor(ssq, m, 32);
  __shared__ float wred[8];
  if (lane == 0) wred[wave] = ssq;
  __syncthreads();
  if (threadIdx.x == 0) {
    float t = 0.f;
    for (int i = 0; i < 8; ++i) t += wred[i];
    ws[WS_SSQ + blockIdx.x] = t;
  }
}

// ---------------- K4: state/tanh/logits + analytic norm --------------------
__global__ __launch_bounds__(1024) void finish_kernel(const float* __restrict__ state,
                                                      const float* __restrict__ Wc,
                                                      const float* __restrict__ bc,
                                                      float* __restrict__ ws,
                                                      float* __restrict__ out) {
  const int tid = threadIdx.x;
  __shared__ float r0[1024], r1[1024], r2[1024];
  float hsh = 0.f, l0 = 0.f, l1 = 0.f;
  for (int c = tid; c < HIDDEN; c += 1024) {
    float yv = 0.f;
#pragma unroll
    for (int ch = 0; ch < 8; ++ch) yv += ws[WS_YPART + ch * HIDDEN + c]; // fixed order
    const float ns = tanhf(yv + state[c]);
    out[2 + c] = ns;                           // new_state
    hsh += yv * ws[WS_HS + c];                 // h' S h
    l0 += ns * Wc[c];
    l1 += ns * Wc[HIDDEN + c];
  }
  r0[tid] = hsh; r1[tid] = l0; r2[tid] = l1;
  __syncthreads();
  for (int s = 512; s; s >>= 1) {
    if (tid < s) { r0[tid] += r0[tid+s]; r1[tid] += r1[tid+s]; r2[tid] += r2[tid+s]; }
    __syncthreads();
  }
  if (tid == 0) {
    float ssqS = 0.f;
    for (int i = 0; i < 512; ++i) ssqS += ws[WS_SSQ + i];   // fixed order
    const float hq = ws[WS_SCAL + 1];
    const float normsq = ssqS + 2.f * HEBB_LR * r0[0] + HEBB_LR * HEBB_LR * hq * hq;
    ws[WS_SCAL + 2] = 1.f / sqrtf(normsq);
    out[0] = r1[0] + bc[0];
    out[1] = r2[0] + bc[1];
  }
}

// ---------------- K5: normalized rank-1 update, NT stream ------------------
// d_out synapses start at float offset 8194 -> only 8B aligned => float2 I/O.
__global__ __launch_bounds__(256) void update_kernel(const float* __restrict__ S,
                                                     const float* __restrict__ ws,
                                                     float* __restrict__ out) {
  const float invn = ws[WS_SCAL + 2];
  const float* __restrict__ hs = ws + WS_HS;
  float* __restrict__ o = out + 2 + HIDDEN;
  const size_t total2 = (size_t)HIDDEN * HIDDEN / 2;   // float2 elements
  const size_t stride = (size_t)gridDim.x * blockDim.x;
  for (size_t i = (size_t)blockIdx.x * blockDim.x + threadIdx.x; i < total2; i += stride) {
    const size_t r = i >> 12;                  // 4096 float2 per row
    const size_t c = (i & 4095) * 2;
    v2f s = __builtin_nontemporal_load((const v2f*)S + i);
    v2f hc = *(const v2f*)(hs + c);
    const float hr = HEBB_LR * hs[r];
    v2f ov;
    ov.x = (s.x + hr * hc.x) * invn;
    ov.y = (s.y + hr * hc.y) * invn;
    __builtin_nontemporal_store(ov, (v2f*)o + i);
  }
}

extern "C" void kernel_launch(void* const* d_in, const int* in_sizes, int n_in,
                              void* d_out, int out_size, void* d_ws, size_t ws_size,
                              hipStream_t stream) {
  const float* x     = (const float*)d_in[0];
  const float* Wenc  = (const float*)d_in[1];
  const float* benc  = (const float*)d_in[2];
  const float* syn   = (const float*)d_in[3];
  const float* state = (const float*)d_in[4];
  const float* Wcls  = (const float*)d_in[5];
  const float* bcls  = (const float*)d_in[6];
  float* out = (float*)d_out;
  float* ws  = (float*)d_ws;

  enc_kernel<<<HIDDEN / 8, 256, 0, stream>>>(x, Wenc, benc, ws);
  topk_kernel<<<1, 1024, 0, stream>>>(ws);
  syn_pass_kernel<<<512, 256, 0, stream>>>(syn, ws);
  finish_kernel<<<1, 1024, 0, stream>>>(state, Wcls, bcls, ws, out);
  update_kernel<<<8192, 256, 0, stream>>>(syn, ws, out);
}